// GraphConv_52106543235734
// MI455X (gfx1250) — compile-verified
//
#include <hip/hip_runtime.h>
#include <hip/hip_bf16.h>

typedef float v2f __attribute__((ext_vector_type(2)));
typedef float v8f __attribute__((ext_vector_type(8)));

#define D 64
#define BN_EPS 1e-5f

// ---------------------------------------------------------------- zero scratch
__global__ void k_zero(float* __restrict__ p, size_t n) {
  size_t i = (size_t)blockIdx.x * blockDim.x + threadIdx.x;
  size_t st = (size_t)gridDim.x * blockDim.x;
  for (; i < n; i += st) p[i] = 0.0f;
}

// ------------------------------------------------------------- edge scatter
// 16 lanes cooperate on one edge; each lane gathers a float4 (16B) of the
// source row and atomically adds it into the per-type accumulator. x (25.6MB)
// and agg (76.8MB) are both L2-resident on MI455X (192MB L2).
__global__ void k_edge_scatter(const float* __restrict__ x,
                               const int* __restrict__ edge_index, // [2][E]
                               const int* __restrict__ edge_attr,  // [E]
                               float* __restrict__ agg,            // [T][N][64]
                               int nEdges, int nNodes) {
  long long g = (long long)blockIdx.x * blockDim.x + threadIdx.x;
  int e = (int)(g >> 4);
  if (e >= nEdges) return;
  int sub = (int)(g & 15);
  int s = edge_index[e];            // src
  int d = edge_index[nEdges + e];   // dst
  int t = edge_attr[e];
  const float4 v = ((const float4*)x)[(size_t)s * (D / 4) + sub];
  float* dst = agg + ((size_t)t * nNodes + (size_t)d) * D + sub * 4;
  unsafeAtomicAdd(dst + 0, v.x);
  unsafeAtomicAdd(dst + 1, v.y);
  unsafeAtomicAdd(dst + 2, v.z);
  unsafeAtomicAdd(dst + 3, v.w);
}

// --------------------------------------------- fused GEMM + bias + ReLU + stats
// out[16,64] = X_tile@W_lin^T + sum_t Agg_t_tile@W_rel[t]^T + bias, per 16-row
// tile, via V_WMMA_F32_16X16X4_F32 (exact f32, matches reference arithmetic).
//
// A frag (16x4 f32): lanes 0-15 hold K=k0,k0+1 ; lanes 16-31 hold K=k0+2,k0+3.
// B frag (4x16 f32): lane n in each half holds column n, K pair per half.
// C/D (16x16 f32):  lane -> N = lane&15 ; VGPR j -> M = j + (lane>=16 ? 8 : 0).
__launch_bounds__(256)
__global__ void k_gemm_relu_stats(const float* __restrict__ x,     // [N][64]
                                  const float* __restrict__ agg,   // [T][N][64]
                                  const float* __restrict__ W_lin, // [64][64]
                                  const float* __restrict__ W_rel, // [T][64][64]
                                  const float* __restrict__ b_lin, // [64]
                                  const float* __restrict__ b_rel, // [T][64]
                                  float* __restrict__ h_out,       // d_out [N][64]
                                  float* __restrict__ gsum,        // [64]
                                  float* __restrict__ gsumsq,      // [64]
                                  int nNodes, int nTypes) {
  extern __shared__ char smem[];
  v2f* ldsB = (v2f*)smem; // [mat(4)][ntile(4)][kstep(16)][lane(32)] = 64KB

  const int tid  = threadIdx.x;
  const int lane = tid & 31;
  const int wave = tid >> 5;
  int mats = nTypes + 1;
  if (mats > 4) mats = 4;

  // Stage all weight B-fragments into LDS (exactly the 64KB of weights).
  for (int idx = tid; idx < mats * 4 * 16 * 32; idx += blockDim.x) {
    int l  = idx & 31;
    int s  = (idx >> 5) & 15;
    int nt = (idx >> 9) & 3;
    int ty = idx >> 11;                   // 0 = W_lin, 1.. = W_rel[ty-1]
    int n  = nt * 16 + (l & 15);          // output feature
    int kb = s * 4 + ((l >> 4) << 1);     // K base for this half-wave
    const float* Wsrc = (ty == 0) ? (W_lin + n * D + kb)
                                  : (W_rel + ((size_t)(ty - 1) * D + n) * D + kb);
    v2f b; b.x = Wsrc[0]; b.y = Wsrc[1];
    ldsB[idx] = b;
  }
  __syncthreads();

  float lsum[4] = {0.f, 0.f, 0.f, 0.f};
  float lsq[4]  = {0.f, 0.f, 0.f, 0.f};

  const int nTiles = (nNodes + 15) >> 4;
  const int wavesInGrid = gridDim.x * (blockDim.x >> 5);

  for (int tile = blockIdx.x * (blockDim.x >> 5) + wave; tile < nTiles;
       tile += wavesInGrid) {
    const int row0 = tile << 4;

    // Accumulators, seeded with total bias (b_lin + sum_t b_rel[t]) per column.
    v8f c[4];
#pragma unroll
    for (int nt = 0; nt < 4; ++nt) {
      int n = nt * 16 + (lane & 15);
      float bi = b_lin[n];
      for (int t = 0; t < nTypes; ++t) bi += b_rel[t * D + n];
#pragma unroll
      for (int j = 0; j < 8; ++j) c[nt][j] = bi;
    }

    int r = row0 + (lane & 15);
    if (r >= nNodes) r = nNodes - 1;      // clamp (harmless duplicate read)
    const int koff = (lane >> 4) << 1;    // 0 or 2

    for (int mat = 0; mat < mats; ++mat) {
      const float* src = (mat == 0) ? x : (agg + (size_t)(mat - 1) * nNodes * D);
      const float* rowp = src + (size_t)r * D + koff;
      const v2f* bbase = ldsB + (size_t)mat * 4 * 16 * 32;
#pragma unroll
      for (int s = 0; s < 16; ++s) {
        v2f a = *(const v2f*)(rowp + s * 4);
#pragma unroll
        for (int nt = 0; nt < 4; ++nt) {
          v2f b = bbase[((nt * 16) + s) * 32 + lane];
          c[nt] = __builtin_amdgcn_wmma_f32_16x16x4_f32(
              false, a, false, b, (short)0, c[nt], false, false);
        }
      }
    }

    // Epilogue: ReLU, store h, accumulate per-feature sum / sum^2 in registers.
    const int mbase = (lane >> 4) << 3;   // 0 or 8
#pragma unroll
    for (int nt = 0; nt < 4; ++nt) {
      const int n = nt * 16 + (lane & 15);
#pragma unroll
      for (int j = 0; j < 8; ++j) {
        int row = row0 + mbase + j;
        float v = c[nt][j];
        v = v > 0.f ? v : 0.f;
        if (row < nNodes) {
          h_out[(size_t)row * D + n] = v;
          lsum[nt] += v;
          lsq[nt]  += v * v;
        }
      }
    }
  }

  // Fold register partials into the global BN statistics.
#pragma unroll
  for (int nt = 0; nt < 4; ++nt) {
    int n = nt * 16 + (lane & 15);
    unsafeAtomicAdd(&gsum[n], lsum[nt]);
    unsafeAtomicAdd(&gsumsq[n], lsq[nt]);
  }
}

// --------------------------------------------------------- BatchNorm finalize
__global__ void k_bn_normalize(float* __restrict__ h,             // [N][64] in place
                               const float* __restrict__ gsum,    // [64]
                               const float* __restrict__ gsumsq,  // [64]
                               const float* __restrict__ gamma,
                               const float* __restrict__ beta,
                               int nNodes) {
  __shared__ float scale[D];
  __shared__ float shift[D];
  if (threadIdx.x < D) {
    int f = threadIdx.x;
    float invN = 1.0f / (float)nNodes;
    float mean = gsum[f] * invN;
    float var  = gsumsq[f] * invN - mean * mean;
    float inv  = rsqrtf(var + BN_EPS);
    float sc   = gamma[f] * inv;
    scale[f] = sc;
    shift[f] = beta[f] - mean * sc;
  }
  __syncthreads();

  size_t total = (size_t)nNodes * (D / 4);
  float4* h4 = (float4*)h;
  size_t i = (size_t)blockIdx.x * blockDim.x + threadIdx.x;
  size_t st = (size_t)gridDim.x * blockDim.x;
  for (; i < total; i += st) {
    float4 v = h4[i];
    int fb = (int)(i & (D / 4 - 1)) * 4;
    v.x = v.x * scale[fb + 0] + shift[fb + 0];
    v.y = v.y * scale[fb + 1] + shift[fb + 1];
    v.z = v.z * scale[fb + 2] + shift[fb + 2];
    v.w = v.w * scale[fb + 3] + shift[fb + 3];
    h4[i] = v;
  }
}

// ---------------------------------------------------------------- launcher
extern "C" void kernel_launch(void* const* d_in, const int* in_sizes, int n_in,
                              void* d_out, int out_size, void* d_ws, size_t ws_size,
                              hipStream_t stream) {
  const float* x          = (const float*)d_in[0];
  const int*   edge_index = (const int*)d_in[1];
  const int*   edge_attr  = (const int*)d_in[2];
  const float* W_rel      = (const float*)d_in[3];
  const float* b_rel      = (const float*)d_in[4];
  const float* W_lin      = (const float*)d_in[5];
  const float* b_lin      = (const float*)d_in[6];
  const float* gamma      = (const float*)d_in[7];
  const float* beta       = (const float*)d_in[8];
  float* out = (float*)d_out;

  const int nNodes = in_sizes[0] / D;
  const int nEdges = in_sizes[1] / 2;
  const int nTypes = in_sizes[3] / (D * D);

  // Workspace layout: [agg: T*N*64 f32][gsum: 64][gsumsq: 64]
  float* agg    = (float*)d_ws;
  size_t aggN   = (size_t)nTypes * nNodes * D;
  float* gsum   = agg + aggN;
  float* gsumsq = gsum + D;

  // 1) zero accumulators + stats (every call: determinism under graph replay)
  {
    size_t n = aggN + 2 * D;
    k_zero<<<4096, 256, 0, stream>>>((float*)d_ws, n);
  }

  // 2) per-type segment-sum scatter over edges
  {
    long long threads = (long long)nEdges * 16;
    int blocks = (int)((threads + 255) / 256);
    k_edge_scatter<<<blocks, 256, 0, stream>>>(x, edge_index, edge_attr, agg,
                                               nEdges, nNodes);
  }

  // 3) fused 4-way GEMM + bias + ReLU + BN statistics (WMMA f32 16x16x4)
  {
    int nTiles = (nNodes + 15) >> 4;
    int blocks = (nTiles + 7) / 8;          // 8 waves per 256-thread block
    if (blocks > 2048) blocks = 2048;
    size_t ldsBytes = 4 * 4 * 16 * 32 * sizeof(v2f); // 64KB weight fragments
    k_gemm_relu_stats<<<blocks, 256, ldsBytes, stream>>>(
        x, agg, W_lin, W_rel, b_lin, b_rel, out, gsum, gsumsq, nNodes, nTypes);
  }

  // 4) BN normalize in place
  {
    size_t total = (size_t)nNodes * (D / 4);
    int blocks = (int)((total + 255) / 256);
    if (blocks > 4096) blocks = 4096;
    k_bn_normalize<<<blocks, 256, 0, stream>>>(out, gsum, gsumsq, gamma, beta,
                                               nNodes);
  }
}